// Downsample_53094385714041
// MI455X (gfx1250) — compile-verified
//
#include <hip/hip_runtime.h>
#include <stdint.h>

// Downsample-by-2 with 4-tap FIR (upfirdn2d, pad=(1,1)) on fp32 NCHW.
// HBM-bound (544 MB moved, ~1 GFLOP) -> stream tiles through LDS with
// CDNA5 async global->LDS copies, compute separable stencil from LDS.

#define H_IN    256
#define W_IN    256
#define H_OUT   128
#define W_OUT   128
#define TILE_OH 16                  // output rows per workgroup
#define ROWS    (TILE_OH * 2 + 2)   // 34 input rows incl. vertical halo

// 16-byte int vector type matching the builtin's signature
typedef int v4i __attribute__((vector_size(16)));
typedef __attribute__((address_space(1))) v4i  g_v4i;   // global
typedef __attribute__((address_space(3))) v4i  l_v4i;   // LDS
typedef __attribute__((address_space(3))) float lds_float;

__device__ __forceinline__ void async_copy_f4(const float* gp, float* lp) {
#if defined(__has_builtin) && __has_builtin(__builtin_amdgcn_global_load_async_to_lds_b128)
    __builtin_amdgcn_global_load_async_to_lds_b128(
        (g_v4i*)gp, (l_v4i*)lp, /*offset=*/0, /*cpol=*/0);
#else
    uint32_t lo = (uint32_t)(uintptr_t)(lds_float*)lp;  // LDS byte offset
    uint64_t ga = (uint64_t)(uintptr_t)gp;
    asm volatile("global_load_async_to_lds_b128 %0, %1, off"
                 :: "v"(lo), "v"(ga) : "memory");
#endif
}

__device__ __forceinline__ void wait_async0() {
#if defined(__has_builtin) && __has_builtin(__builtin_amdgcn_s_wait_asynccnt)
    __builtin_amdgcn_s_wait_asynccnt(0);
#else
    asm volatile("s_wait_asynccnt 0" ::: "memory");
#endif
}

__global__ __launch_bounds__(256)
void downsample2x_fir4(const float* __restrict__ x,
                       const float* __restrict__ kern,
                       float* __restrict__ out) {
    __shared__ float smem[ROWS * W_IN];   // 34 KB

    const int tid     = threadIdx.x;      // 0..255
    const int plane   = blockIdx.x;       // b*C + c
    const int tile    = blockIdx.y;       // 0..H_OUT/TILE_OH-1
    const int oh0     = tile * TILE_OH;
    const int in_row0 = oh0 * 2 - 1;      // first input row (may be -1)

    const float* src = x + (size_t)plane * (H_IN * W_IN);

    // ---- stage ROWS x 256 fp32 into LDS via async b128 copies ----
    // 64 threads (2 waves) per row -> row index is wave-uniform, EXEC full.
    {
        const int r_start = tid >> 6;         // 0..3
        const int c4      = (tid & 63) * 4;   // 16B chunk start column
        for (int r = r_start; r < ROWS; r += 4) {
            const int in_row = in_row0 + r;
            if (in_row >= 0 && in_row < H_IN) {
                async_copy_f4(src + (size_t)in_row * W_IN + c4,
                              &smem[r * W_IN + c4]);
            } else {                           // vertical zero-pad halo
                smem[r * W_IN + c4 + 0] = 0.f;
                smem[r * W_IN + c4 + 1] = 0.f;
                smem[r * W_IN + c4 + 2] = 0.f;
                smem[r * W_IN + c4 + 3] = 0.f;
            }
        }
    }

    // ---- separable taps from the (flipped) rank-1 4x4 kernel ----
    // Reference convolves with kernel[::-1,::-1]; w[i][j] = a_i * b_j.
    float w[16];
    #pragma unroll
    for (int i = 0; i < 4; ++i)
        #pragma unroll
        for (int j = 0; j < 4; ++j)
            w[i * 4 + j] = kern[(3 - i) * 4 + (3 - j)];
    const float va0 = w[0]  + w[1]  + w[2]  + w[3];    // a_i * sum(b)
    const float va1 = w[4]  + w[5]  + w[6]  + w[7];
    const float va2 = w[8]  + w[9]  + w[10] + w[11];
    const float va3 = w[12] + w[13] + w[14] + w[15];
    const float inv = 1.0f / va0;                      // hb_j = b_j / sum(b)
    const float hb0 = w[0] * inv, hb1 = w[1] * inv;
    const float hb2 = w[2] * inv, hb3 = w[3] * inv;
    // va_i * hb_j == w[i][j] exactly for the rank-1 FIR kernel.

    wait_async0();        // per-wave: all async LDS writes landed
    __syncthreads();      // workgroup-visible

    // ---- compute: lane = output column (conflict-free LDS), 8 rows/thread
    const int ocol  = tid & 127;          // 0..127
    const int g     = tid >> 7;           // 0 or 1 (row half)
    const int lr0   = g * (TILE_OH / 2);  // first local output row
    const int tr0   = lr0 * 2;            // first LDS tile row (18 used)
    const int cbase = ocol * 2 - 1;       // leftmost input column (may be -1)

    float h[18];                           // horizontal pass
    #pragma unroll
    for (int r = 0; r < 18; ++r) {
        const float* row = &smem[(tr0 + r) * W_IN];
        const float x0 = (cbase >= 0)         ? row[cbase]     : 0.f;
        const float x1 =                        row[cbase + 1];
        const float x2 =                        row[cbase + 2];
        const float x3 = (cbase + 3 < W_IN)   ? row[cbase + 3] : 0.f;
        h[r] = hb0 * x0 + hb1 * x1 + hb2 * x2 + hb3 * x3;
    }

    float* dst = out + (size_t)plane * (H_OUT * W_OUT)
                     + (size_t)(oh0 + lr0) * W_OUT + ocol;
    #pragma unroll
    for (int j = 0; j < 8; ++j) {          // vertical pass + store
        const float o = va0 * h[2 * j]     + va1 * h[2 * j + 1]
                      + va2 * h[2 * j + 2] + va3 * h[2 * j + 3];
        dst[(size_t)j * W_OUT] = o;
    }
}

extern "C" void kernel_launch(void* const* d_in, const int* in_sizes, int n_in,
                              void* d_out, int out_size, void* d_ws, size_t ws_size,
                              hipStream_t stream) {
    (void)n_in; (void)out_size; (void)d_ws; (void)ws_size;
    const float* x    = (const float*)d_in[0];   // (8,256,256,256) fp32
    const float* kern = (const float*)d_in[1];   // (4,4) fp32
    float*       out  = (float*)d_out;           // (8,256,128,128) fp32

    const int planes = in_sizes[0] / (H_IN * W_IN);   // B*C = 2048
    dim3 grid(planes, H_OUT / TILE_OH);               // (2048, 8)
    downsample2x_fir4<<<grid, 256, 0, stream>>>(x, kern, out);
}